// TriggerAwareGAT_55396488184506
// MI455X (gfx1250) — compile-verified
//
#include <hip/hip_runtime.h>
#include <math.h>

#define NN 4096
#define HD 128
#define NH 4
#define DH 32

typedef __attribute__((ext_vector_type(2))) float v2f;
typedef __attribute__((ext_vector_type(8))) float v8f;

#if (__has_builtin(__builtin_amdgcn_global_load_async_to_lds_b128) && \
     __has_builtin(__builtin_amdgcn_global_load_async_to_lds_b64))
#define ASYNC_OK 1
#else
#define ASYNC_OK 0
#endif

#if defined(__AMDGCN__)
typedef int v4i_ __attribute__((vector_size(16)));
typedef int v2i_ __attribute__((vector_size(8)));
typedef __attribute__((address_space(1))) v4i_ g_v4i;
typedef __attribute__((address_space(3))) v4i_ s_v4i;
typedef __attribute__((address_space(1))) v2i_ g_v2i;
typedef __attribute__((address_space(3))) v2i_ s_v2i;
#endif

#if ASYNC_OK
#define GLD_B128(gsrc, ldst) \
  __builtin_amdgcn_global_load_async_to_lds_b128((g_v4i*)(gsrc), (s_v4i*)(ldst), 0, 0)
#define GLD_B64(gsrc, ldst) \
  __builtin_amdgcn_global_load_async_to_lds_b64((g_v2i*)(gsrc), (s_v2i*)(ldst), 0, 0)
#if __has_builtin(__builtin_amdgcn_s_wait_asynccnt)
#define WAIT_ASYNC(n) __builtin_amdgcn_s_wait_asynccnt(n)
#else
#define WAIT_ASYNC(n) asm volatile("s_wait_asynccnt %0" :: "n"(n) : "memory")
#endif
#endif

// ---------------- K0: trigger-derived scalars ----------------
// scal[0..3] = 1/(sqrt(32)*clip(temp)), scal[4] = gate, scal[5..8] = stb*is_reactive
__global__ void k_scalars(const float* __restrict__ trig,
                          const float* __restrict__ Wt1, const float* __restrict__ bt1,
                          const float* __restrict__ Wt2, const float* __restrict__ bt2,
                          const float* __restrict__ Wg,  const float* __restrict__ bg,
                          const float* __restrict__ stb, float* __restrict__ scal) {
  __shared__ float t1[64];
  __shared__ float gv[128];
  const int t = threadIdx.x; // 128 threads
  const float tx = trig[0], ty = trig[1], tz = trig[2];
  if (t < 64) {
    float z = Wt1[t*3]*tx + Wt1[t*3+1]*ty + Wt1[t*3+2]*tz + bt1[t];
    t1[t] = fmaxf(z, 0.0f);
  }
  {
    float z = Wg[t*3]*tx + Wg[t*3+1]*ty + Wg[t*3+2]*tz + bg[t];
    gv[t] = 1.0f / (1.0f + __expf(-z));
  }
  __syncthreads();
  if (t < 4) {
    float z = bt2[t];
    for (int k = 0; k < 64; ++k) z += Wt2[t*64+k]*t1[k];
    float sp = (z > 20.0f) ? z : log1pf(__expf(z));   // softplus
    sp = fminf(fmaxf(sp, 0.1f), 5.0f);
    scal[t]   = 1.0f / (sp * sqrtf((float)DH));
    scal[5+t] = stb[t] * (1.0f - tx);
  }
  if (t == 0) {
    float s = 0.0f;
    for (int k = 0; k < 128; ++k) s += gv[k];
    scal[4] = s * (1.0f/128.0f);
  }
}

// ---------------- K1: h = relu(LN(enriched @ Wp.T + bp)); also sf[i]=nf[i,2] ----------------
__global__ void k_hidden(const float* __restrict__ nf, const float* __restrict__ trig,
                         const float* __restrict__ Wp, const float* __restrict__ bp,
                         const float* __restrict__ g,  const float* __restrict__ b,
                         float* __restrict__ h, float* __restrict__ sf) {
  __shared__ float red[128];
  __shared__ float sh[2];
  const int i = blockIdx.x, c = threadIdx.x;
  const float e0 = nf[i*3], e1 = nf[i*3+1], e2 = nf[i*3+2];
  const float t0 = trig[0], t1 = trig[1], t2 = trig[2];
  if (c == 0) sf[i] = e2;
  const float* w = Wp + c*6;
  float p = w[0]*e0 + w[1]*e1 + w[2]*e2 + w[3]*t0 + w[4]*t1 + w[5]*t2 + bp[c];
  red[c] = p; __syncthreads();
  for (int s = 64; s > 0; s >>= 1) { if (c < s) red[c] += red[c+s]; __syncthreads(); }
  if (c == 0) sh[0] = red[0] * (1.0f/128.0f);
  __syncthreads();
  const float mu = sh[0];
  const float d  = p - mu;
  red[c] = d*d; __syncthreads();
  for (int s = 64; s > 0; s >>= 1) { if (c < s) red[c] += red[c+s]; __syncthreads(); }
  if (c == 0) sh[1] = red[0] * (1.0f/128.0f);
  __syncthreads();
  const float y = d * rsqrtf(sh[1] + 1e-5f) * g[c] + b[c];
  h[i*HD + c] = fmaxf(y, 0.0f);
}

// ---------------- K2: Q,K,V GEMMs (Q pre-scaled per head) ----------------
__global__ void k_qkv(const float* __restrict__ h,
                      const float* __restrict__ Wq, const float* __restrict__ Wk,
                      const float* __restrict__ Wv, const float* __restrict__ scal,
                      float* __restrict__ Qs, float* __restrict__ Kd, float* __restrict__ Vd) {
  __shared__ float hr[128];
  const int i = blockIdx.x, t = threadIdx.x; // 384 threads
  if (t < 128) hr[t] = h[i*HD + t];
  __syncthreads();
  const float* W; float* out; int c;
  if      (t < 128) { c = t;       W = Wq + c*HD; out = Qs; }
  else if (t < 256) { c = t - 128; W = Wk + c*HD; out = Kd; }
  else              { c = t - 256; W = Wv + c*HD; out = Vd; }
  float acc = 0.0f;
  for (int k = 0; k < HD; ++k) acc += W[k]*hr[k];
  if (t < 128) acc *= scal[c >> 5];       // fold 1/(sqrt(dh)*temp[head]) into Q
  out[i*HD + c] = acc;
}

// ---------------- K3: fused flash-style attention (WMMA f32 16x16x4) ----------------
// grid = N/16 blocks of 128 threads; wave w handles head w for the same 16-row tile.
// Async path: K/V tiles double-buffered, adj tiles 4-deep, via GLOBAL_LOAD_ASYNC_TO_LDS.
__global__ void __launch_bounds__(128)
k_attn(const float* __restrict__ Qs, const float* __restrict__ Kd,
       const float* __restrict__ Vd, const float* __restrict__ adj,
       const float* __restrict__ sf, const float* __restrict__ scal,
       float* __restrict__ agg) {
#if ASYNC_OK
  __shared__ __align__(16) float KT[2][16][HD];
  __shared__ __align__(16) float VT[2][16][HD];
  __shared__ __align__(16) float AT[4][16][16];
#else
  __shared__ __align__(16) float KT[1][16][HD];
  __shared__ __align__(16) float VT[1][16][HD];
  __shared__ __align__(16) float AT[1][16][16];
#endif
  __shared__ float PB[NH][16][16];

  const int t     = threadIdx.x;
  const int wave  = t >> 5;          // head index
  const int lane  = t & 31;
  const int l16   = lane & 15;
  const int half  = lane >> 4;
  const int i0    = blockIdx.x * 16;
  const int hbase = wave * DH;

  // Q A-fragments: A[lane][v] = Q[i0+l16][hbase + 4*c + v + 2*half]
  float qa0[8], qa1[8];
  {
    const float* qrow = Qs + (size_t)(i0 + l16)*HD + hbase + 2*half;
    #pragma unroll
    for (int c = 0; c < 8; ++c) { qa0[c] = qrow[4*c]; qa1[c] = qrow[4*c + 1]; }
  }
  // per-row stateful values for this lane's 8 rows
  float sfr[8];
  #pragma unroll
  for (int r = 0; r < 8; ++r) sfr[r] = sf[i0 + r + 8*half];

  const float stbr = scal[5 + wave];
  const float gate = scal[4];

  v8f accLo = {0.f,0.f,0.f,0.f,0.f,0.f,0.f,0.f};
  v8f accHi = {0.f,0.f,0.f,0.f,0.f,0.f,0.f,0.f};
  float sump[8] = {0.f,0.f,0.f,0.f,0.f,0.f,0.f,0.f};

  // cooperative-load index decomposition (shared by both paths)
  const int adr = t >> 3;            // adj row 0..15
  const int adc = (t & 7) * 2;       // adj col pair

#if ASYNC_OK
  // --- async tile issue helpers (9 ops/wave/tile: 4xK b128, 4xV b128, 1 adj b64) ---
  #define ISSUE_KV(J0, BUF)                                                          \
    do {                                                                             \
      _Pragma("unroll")                                                              \
      for (int q = 0; q < 4; ++q) {                                                  \
        const int p = t + 128*q; const int row = p >> 5; const int c4 = p & 31;      \
        GLD_B128(((const float4*)(Kd + (size_t)((J0)+row)*HD)) + c4,                 \
                 ((float4*)&KT[BUF][row][0]) + c4);                                  \
      }                                                                              \
      _Pragma("unroll")                                                              \
      for (int q = 0; q < 4; ++q) {                                                  \
        const int p = t + 128*q; const int row = p >> 5; const int c4 = p & 31;      \
        GLD_B128(((const float4*)(Vd + (size_t)((J0)+row)*HD)) + c4,                 \
                 ((float4*)&VT[BUF][row][0]) + c4);                                  \
      }                                                                              \
    } while (0)
  #define ISSUE_ADJ(J0, BUF)                                                         \
    GLD_B64(adj + (size_t)(i0 + adr)*NN + (J0) + adc, &AT[BUF][adr][adc])

  // prologue: tile 0 K/V, adj tiles 0..2
  ISSUE_KV(0, 0);
  ISSUE_ADJ(0, 0);
  ISSUE_ADJ(16, 1);
  ISSUE_ADJ(32, 2);
#endif

  float sjv_next = sf[l16];

  for (int jt = 0; jt < NN/16; ++jt) {
    const int j0 = jt * 16;
#if ASYNC_OK
    const int kvb = jt & 1;
    const int ab  = jt & 3;
    if (jt + 1 < NN/16) ISSUE_KV((jt+1)*16, kvb ^ 1);
    if (jt + 3 < NN/16) ISSUE_ADJ((jt+3)*16, (jt+3) & 3);
    // in-order async completion: exactly 10 ops are younger than the last op of tile jt
    if (jt < NN/16 - 3) { WAIT_ASYNC(10); } else { WAIT_ASYNC(0); }
    __syncthreads();
#else
    const int kvb = 0, ab = 0;
    __syncthreads();
    #pragma unroll
    for (int q = 0; q < 4; ++q) {
      const int p   = t + 128*q;
      const int row = p >> 5;
      const int c4  = p & 31;
      ((float4*)&KT[0][row][0])[c4] = ((const float4*)(Kd + (size_t)(j0+row)*HD))[c4];
      ((float4*)&VT[0][row][0])[c4] = ((const float4*)(Vd + (size_t)(j0+row)*HD))[c4];
    }
    {
      const float* arow = adj + (size_t)(i0 + adr)*NN + j0 + adc;
      AT[0][adr][adc]   = arow[0];
      AT[0][adr][adc+1] = arow[1];
      if (j0 + 272 <= NN) __builtin_prefetch(arow + 256, 0, 0);
    }
    __syncthreads();
#endif

    const float sjv = sjv_next;
    if (jt + 1 < NN/16) sjv_next = sf[j0 + 16 + l16];

    // --- S = Qs @ K^T (K-dim 32 -> 8x wmma_f32_16x16x4) ---
    v8f s = {0.f,0.f,0.f,0.f,0.f,0.f,0.f,0.f};
    #pragma unroll
    for (int c = 0; c < 8; ++c) {
      v2f a; a.x = qa0[c]; a.y = qa1[c];
      v2f b; b.x = KT[kvb][l16][hbase + 4*c + 2*half];
             b.y = KT[kvb][l16][hbase + 4*c + 1 + 2*half];
      s = __builtin_amdgcn_wmma_f32_16x16x4_f32(false, a, false, b, (short)0, s, false, false);
    }

    // --- mask, stateful bias, exp; stage P into LDS (C-layout -> A-layout bridge) ---
    #pragma unroll
    for (int r = 0; r < 8; ++r) {
      const int row = r + 8*half;
      float sv = s[r];
      const float av = AT[ab][row][l16];
      sv = (av == 0.0f) ? -1.0e9f : sv;           // mask BEFORE bias (matches reference)
      sv += sfr[r] * sjv * stbr;
      const float pe = __expf(sv);                // max-free: scores bounded, exp(-1e9)->0
      sump[r] += pe;
      PB[wave][row][l16] = pe;
    }

    // --- D += P @ V (K-dim 16 -> 4 chunks; two 16x16 output tiles d<16, d>=16) ---
    #pragma unroll
    for (int q = 0; q < 4; ++q) {
      const int jb = 4*q + 2*half;
      v2f a;  a.x  = PB[wave][l16][jb];       a.y  = PB[wave][l16][jb+1];
      v2f bl; bl.x = VT[kvb][jb][hbase + l16];      bl.y = VT[kvb][jb+1][hbase + l16];
      v2f bh; bh.x = VT[kvb][jb][hbase + 16 + l16]; bh.y = VT[kvb][jb+1][hbase + 16 + l16];
      accLo = __builtin_amdgcn_wmma_f32_16x16x4_f32(false, a, false, bl, (short)0, accLo, false, false);
      accHi = __builtin_amdgcn_wmma_f32_16x16x4_f32(false, a, false, bh, (short)0, accHi, false, false);
    }
    __syncthreads();   // all waves done reading this KV/adj buffer before it is refilled
  }

  // --- row-sum reduction across the 16-lane group holding each row ---
  #pragma unroll
  for (int r = 0; r < 8; ++r) {
    float v = sump[r];
    v += __shfl_xor(v, 1, 32);
    v += __shfl_xor(v, 2, 32);
    v += __shfl_xor(v, 4, 32);
    v += __shfl_xor(v, 8, 32);
    sump[r] = v;
  }

  // --- agg = gate * D / rowsum ---
  #pragma unroll
  for (int r = 0; r < 8; ++r) {
    const int row = r + 8*half;
    const float sc = gate / sump[r];
    float* orow = agg + (size_t)(i0 + row)*HD + hbase;
    orow[l16]      = accLo[r] * sc;
    orow[16 + l16] = accHi[r] * sc;
  }
}

// ---------------- K4: out = LN(agg @ Wo.T + bo) + res * h ----------------
__global__ void k_out(const float* __restrict__ agg, const float* __restrict__ hbuf,
                      const float* __restrict__ Wo,  const float* __restrict__ bo,
                      const float* __restrict__ g,   const float* __restrict__ b,
                      const float* __restrict__ res, float* __restrict__ out) {
  __shared__ float ar[128];
  __shared__ float red[128];
  __shared__ float sh[2];
  const int i = blockIdx.x, c = threadIdx.x;
  ar[c] = agg[i*HD + c];
  __syncthreads();
  const float* w = Wo + c*HD;
  float acc = bo[c];
  for (int k = 0; k < HD; ++k) acc += w[k]*ar[k];
  red[c] = acc; __syncthreads();
  for (int s = 64; s > 0; s >>= 1) { if (c < s) red[c] += red[c+s]; __syncthreads(); }
  if (c == 0) sh[0] = red[0] * (1.0f/128.0f);
  __syncthreads();
  const float mu = sh[0];
  const float d  = acc - mu;
  red[c] = d*d; __syncthreads();
  for (int s = 64; s > 0; s >>= 1) { if (c < s) red[c] += red[c+s]; __syncthreads(); }
  if (c == 0) sh[1] = red[0] * (1.0f/128.0f);
  __syncthreads();
  const float y = d * rsqrtf(sh[1] + 1e-5f) * g[c] + b[c];
  out[i*HD + c] = y + res[0]*hbuf[i*HD + c];
}

extern "C" void kernel_launch(void* const* d_in, const int* in_sizes, int n_in,
                              void* d_out, int out_size, void* d_ws, size_t ws_size,
                              hipStream_t stream) {
  const float* nf   = (const float*)d_in[0];
  const float* adj  = (const float*)d_in[1];
  const float* trig = (const float*)d_in[2];
  const float* Wp   = (const float*)d_in[3];
  const float* bp   = (const float*)d_in[4];
  const float* ln1g = (const float*)d_in[5];
  const float* ln1b = (const float*)d_in[6];
  const float* Wq   = (const float*)d_in[7];
  const float* Wk   = (const float*)d_in[8];
  const float* Wv   = (const float*)d_in[9];
  const float* Wt1  = (const float*)d_in[10];
  const float* bt1  = (const float*)d_in[11];
  const float* Wt2  = (const float*)d_in[12];
  const float* bt2  = (const float*)d_in[13];
  const float* Wg   = (const float*)d_in[14];
  const float* bg   = (const float*)d_in[15];
  const float* stb  = (const float*)d_in[16];
  const float* Wo   = (const float*)d_in[17];
  const float* bo   = (const float*)d_in[18];
  const float* ln2g = (const float*)d_in[19];
  const float* ln2b = (const float*)d_in[20];
  const float* res  = (const float*)d_in[21];
  float* out = (float*)d_out;

  float* ws   = (float*)d_ws;
  float* h    = ws;                 // N*HD
  float* Qs   = ws + 1*NN*HD;       // N*HD (pre-scaled)
  float* Kd   = ws + 2*NN*HD;       // N*HD
  float* Vd   = ws + 3*NN*HD;       // N*HD
  float* agg  = ws + 4*NN*HD;       // N*HD
  float* scal = ws + 5*NN*HD;       // 9 floats (padded to 16)
  float* sfv  = ws + 5*NN*HD + 16;  // N floats (stateful column)

  k_scalars<<<1, 128, 0, stream>>>(trig, Wt1, bt1, Wt2, bt2, Wg, bg, stb, scal);
  k_hidden <<<NN, 128, 0, stream>>>(nf, trig, Wp, bp, ln1g, ln1b, h, sfv);
  k_qkv    <<<NN, 384, 0, stream>>>(h, Wq, Wk, Wv, scal, Qs, Kd, Vd);
  k_attn   <<<NN/16, 128, 0, stream>>>(Qs, Kd, Vd, adj, sfv, scal, agg);
  k_out    <<<NN, 128, 0, stream>>>(agg, h, Wo, bo, ln2g, ln2b, res, out);
}